// Prototype_Balanced_Contrastive_Loss_33414845562974
// MI455X (gfx1250) — compile-verified
//
#include <hip/hip_runtime.h>
#include <hip/hip_bf16.h>

// ---------------- problem constants (from setup_inputs) ----------------
#define B_     8
#define C_     256            // feature dim == blockDim for reductions
#define HW_    16384          // 128*128
#define W_     128
#define W0_    512
#define H0W0   262144         // 512*512
#define NCP1   21             // num_class + 1
#define SLOTS  9              // B + 1
#define RROWS  189            // NCP1 * SLOTS
#define RP     192            // RROWS padded to multiple of 16
#define INV_T  10.0f          // 1 / temperature(0.1)
#define EPSN   1e-12f

#define PIXC   1024           // pixels per segment-sum block
#define NCHUNK 16             // HW_/PIXC
#define ACCS   257            // padded class stride in LDS accumulator

typedef float v2f __attribute__((ext_vector_type(2)));
typedef float v8f __attribute__((ext_vector_type(8)));

// ---------------- workspace layout (bytes) ----------------
#define OFF_SUMS_T   0u
#define OFF_SUMS_S   172032u                 // 8*21*256*4
#define OFF_CNTPIX_T 344064u
#define OFF_CNTPIX_S 345088u
#define OFF_P_T      346112u                 // 192*256*4 each
#define OFF_P_S      542720u
#define OFF_CNT_T    739328u
#define OFF_CNT_S    739584u
#define OFF_GSS      739840u                 // 192*192*4 each
#define OFF_GST      887296u
#define ZERO_WORDS   (739840u / 4u)          // zero everything before the Grams

// ---------------- kernels ----------------
__global__ __launch_bounds__(256) void zero_ws_k(float* p, unsigned n) {
    unsigned i = blockIdx.x * 256u + threadIdx.x;
    unsigned stride = gridDim.x * 256u;
    for (; i < n; i += stride) p[i] = 0.0f;
}

// Streaming label-segmented sum: sums[b][c][d] += feat[b][d][p] where c = lab4x(p).
// One block = (batch b, pixel chunk of PIXC). LDS accumulate, then global f32 atomics.
__global__ __launch_bounds__(256) void seg_sums_k(const int* __restrict__ lab,
                                                  const float* __restrict__ feat,
                                                  float* __restrict__ sums,
                                                  int* __restrict__ cntpix) {
    __shared__ int   slab[PIXC];
    __shared__ float acc[NCP1 * ACCS];
    __shared__ int   hist[NCP1];

    const int tid = threadIdx.x;
    const int chunk = blockIdx.x;
    const int b = blockIdx.y;
    const int pixbase = chunk * PIXC;

    for (int i = tid; i < NCP1 * ACCS; i += 256) acc[i] = 0.0f;
    if (tid < NCP1) hist[tid] = 0;
    __syncthreads();

    // nearest-down labels: lab[b,0,4y,4x]
    for (int i = tid; i < PIXC; i += 256) {
        int p = pixbase + i;
        int y = p >> 7, x = p & (W_ - 1);
        int l = lab[b * H0W0 + (y << 2) * W0_ + (x << 2)];
        l = l < 0 ? 0 : (l >= NCP1 ? NCP1 - 1 : l);
        slab[i] = l;
        atomicAdd(&hist[l], 1);
    }
    __syncthreads();

    int myLab[4];
#pragma unroll
    for (int j = 0; j < 4; ++j) myLab[j] = slab[tid * 4 + j];

    const float* fbase = feat + (size_t)b * C_ * HW_ + pixbase;
    for (int d = 0; d < C_; ++d) {
        const float4 v = ((const float4*)(fbase + (size_t)d * HW_))[tid]; // b128 coalesced
        atomicAdd(&acc[myLab[0] * ACCS + d], v.x);   // ds_add_f32
        atomicAdd(&acc[myLab[1] * ACCS + d], v.y);
        atomicAdd(&acc[myLab[2] * ACCS + d], v.z);
        atomicAdd(&acc[myLab[3] * ACCS + d], v.w);
    }
    __syncthreads();

    for (int i = tid; i < NCP1 * C_; i += 256) {
        int c = i >> 8, d = i & (C_ - 1);
        float v = acc[c * ACCS + d];
        if (v != 0.0f) atomicAdd(&sums[((size_t)b * NCP1 + c) * C_ + d], v);
    }
    if (tid < NCP1 && hist[tid] > 0) atomicAdd(&cntpix[b * NCP1 + tid], hist[tid]);
}

// Build P[class][slot][d]: per-batch normalized class means (slot-compacted),
// then the normalized global prototype appended at slot cnt_b; cnt = cnt_b (+1 for c>=1).
__global__ __launch_bounds__(256) void build_protos_k(const float* __restrict__ sums,
                                                      const int* __restrict__ cntpix,
                                                      const float* __restrict__ protos,
                                                      float* __restrict__ P,
                                                      int* __restrict__ cntArr) {
    const int c = blockIdx.x;
    const int tid = threadIdx.x;     // tid == feature dim d (C_ == 256)
    __shared__ float red[256];
    __shared__ float s_norm;

    int slot = 0;
    for (int b = 0; b < B_; ++b) {
        int cnt = cntpix[b * NCP1 + c];
        bool present = (cnt > 0) && (c != 0);      // uniform across block
        if (present) {
            float mean = sums[((size_t)b * NCP1 + c) * C_ + tid] / (float)cnt;
            red[tid] = mean * mean;
            __syncthreads();
            for (int st = 128; st > 0; st >>= 1) {
                if (tid < st) red[tid] += red[tid + st];
                __syncthreads();
            }
            if (tid == 0) s_norm = sqrtf(red[0]);
            __syncthreads();
            P[((size_t)c * SLOTS + slot) * C_ + tid] = mean / fmaxf(s_norm, EPSN);
            ++slot;
            __syncthreads();
        }
    }
    if (c >= 1) {
        float v = protos[c * C_ + tid];
        red[tid] = v * v;
        __syncthreads();
        for (int st = 128; st > 0; st >>= 1) {
            if (tid < st) red[tid] += red[tid + st];
            __syncthreads();
        }
        if (tid == 0) s_norm = sqrtf(red[0]);
        __syncthreads();
        P[((size_t)c * SLOTS + slot) * C_ + tid] = v / fmaxf(s_norm, EPSN);
        if (tid == 0) cntArr[c] = slot + 1;
    } else {
        if (tid == 0) cntArr[c] = 0;
    }
}

// Gram matrices via V_WMMA_F32_16X16X4_F32. One wave per 16x16 output tile.
// A-tile 16x4 (f32): lane L holds A[M=L%16][K0=(L/16)*2 .. +1]  -> float2 along K.
// B-tile 4x16 = rows of the other matrix transposed: same per-lane float2 pattern.
__global__ __launch_bounds__(32) void gram_k(const float* __restrict__ Ps,
                                             const float* __restrict__ Pt,
                                             float* __restrict__ Gss,
                                             float* __restrict__ Gst) {
    const int tm = blockIdx.x, tn = blockIdx.y;
    const float* Bm = (blockIdx.z == 0) ? Ps : Pt;
    float* out = (blockIdx.z == 0) ? Gss : Gst;

    const int lane = threadIdx.x;
    const int mrow = tm * 16 + (lane & 15);
    const int nrow = tn * 16 + (lane & 15);
    const int koff = (lane >> 4) * 2;

    const float* arow = Ps + (size_t)mrow * C_ + koff;
    const float* brow = Bm + (size_t)nrow * C_ + koff;

    v8f acc = {};
    for (int k = 0; k < C_; k += 4) {
        v2f a = *(const v2f*)(arow + k);
        v2f b = *(const v2f*)(brow + k);
        acc = __builtin_amdgcn_wmma_f32_16x16x4_f32(
            /*neg_a=*/false, a, /*neg_b=*/false, b,
            /*c_mod=*/(short)0, acc, /*reuse_a=*/false, /*reuse_b=*/false);
    }
#pragma unroll
    for (int v = 0; v < 8; ++v) {
        int row = tm * 16 + v + (lane >> 4) * 8;  // C/D layout: M = vgpr + (lane/16)*8
        int col = tn * 16 + (lane & 15);          // N = lane%16
        out[(size_t)row * RP + col] = acc[v];
    }
}

// Final scalar loss. Deterministic fixed-order LDS reductions.
__global__ __launch_bounds__(256) void loss_k(const float* __restrict__ Gss,
                                              const float* __restrict__ Gst,
                                              const int* __restrict__ cnt_s,
                                              const int* __restrict__ cnt_t,
                                              const int* __restrict__ noldp,
                                              float* __restrict__ out) {
    __shared__ float ld[RROWS];
    __shared__ int cs[NCP1], ct[NCP1];
    __shared__ float redA[256], redB[256];
    const int tid = threadIdx.x;

    if (tid < NCP1) { cs[tid] = cnt_s[tid]; ct[tid] = cnt_t[tid]; }
    __syncthreads();

    // log-denominator per anchor row r = i*9+s
    for (int r = tid; r < RROWS; r += 256) {
        const float* row = Gss + (size_t)r * RP;
        float denom = 0.0f;
        for (int m = 1; m < NCP1; ++m) {
            int cm = cs[m];
            float sm = 0.0f;
            for (int n = 0; n < cm; ++n) sm += expf(row[m * SLOTS + n] * INV_T);
            denom += sm / (float)(cm > 0 ? cm : 1);
        }
        ld[r] = logf(denom);
    }
    __syncthreads();

    const int NOLD = noldp[0];
    float aOld = 0.0f, aNew = 0.0f;
    for (int e = tid; e < NCP1 * 81; e += 256) {
        int i = e / 81, s = (e / 9) % 9, k = e % 9;
        int ci = cs[i];
        if (i >= 1 && ci > 1 && s < ci - 1) {
            int r = i * SLOTS + s;
            if (i <= NOLD) {
                int cti = ct[i];
                if (k < cti) {
                    float term = -Gst[(size_t)r * RP + i * SLOTS + k] * INV_T + ld[r];
                    aOld += term / ((float)cti * (float)(ci - 1));
                }
            } else {
                if (k < ci && k != s) {
                    float term = -Gss[(size_t)r * RP + i * SLOTS + k] * INV_T + ld[r];
                    aNew += term / ((float)(ci - 1) * (float)(ci - 1));
                }
            }
        }
    }
    redA[tid] = aOld; redB[tid] = aNew;
    __syncthreads();
    for (int st = 128; st > 0; st >>= 1) {
        if (tid < st) { redA[tid] += redA[tid + st]; redB[tid] += redB[tid + st]; }
        __syncthreads();
    }
    if (tid == 0) {
        int nO = 0, nN = 0;
        for (int i = 1; i < NCP1; ++i)
            if (cs[i] > 1) { if (i <= NOLD) ++nO; else ++nN; }
        float lO = (nO > 0) ? redA[0] / (float)nO : 0.0f;
        float lN = (nN > 0) ? redB[0] / (float)nN : 0.0f;
        out[0] = lO + 0.1f * lN;
    }
}

// ---------------- host ----------------
extern "C" void kernel_launch(void* const* d_in, const int* in_sizes, int n_in,
                              void* d_out, int out_size, void* d_ws, size_t ws_size,
                              hipStream_t stream) {
    const int*   lab_t  = (const int*)d_in[0];   // pseudo_label_old_down
    const int*   lab_s  = (const int*)d_in[1];   // pseudo_label
    const float* feat_t = (const float*)d_in[2]; // features_old
    const float* feat_s = (const float*)d_in[3]; // features
    const float* protos = (const float*)d_in[4]; // prototypes
    const int*   nold   = (const int*)d_in[6];   // num_old_class (device scalar)
    float*       outp   = (float*)d_out;

    char* ws = (char*)d_ws;
    float* sums_t   = (float*)(ws + OFF_SUMS_T);
    float* sums_s   = (float*)(ws + OFF_SUMS_S);
    int*   cntpix_t = (int*)  (ws + OFF_CNTPIX_T);
    int*   cntpix_s = (int*)  (ws + OFF_CNTPIX_S);
    float* P_t      = (float*)(ws + OFF_P_T);
    float* P_s      = (float*)(ws + OFF_P_S);
    int*   cnt_t    = (int*)  (ws + OFF_CNT_T);
    int*   cnt_s    = (int*)  (ws + OFF_CNT_S);
    float* Gss      = (float*)(ws + OFF_GSS);
    float* Gst      = (float*)(ws + OFF_GST);

    zero_ws_k<<<256, 256, 0, stream>>>((float*)ws, ZERO_WORDS);

    seg_sums_k<<<dim3(NCHUNK, B_), 256, 0, stream>>>(lab_t, feat_t, sums_t, cntpix_t);
    seg_sums_k<<<dim3(NCHUNK, B_), 256, 0, stream>>>(lab_s, feat_s, sums_s, cntpix_s);

    build_protos_k<<<NCP1, 256, 0, stream>>>(sums_t, cntpix_t, protos, P_t, cnt_t);
    build_protos_k<<<NCP1, 256, 0, stream>>>(sums_s, cntpix_s, protos, P_s, cnt_s);

    gram_k<<<dim3(RP / 16, RP / 16, 2), 32, 0, stream>>>(P_s, P_t, Gss, Gst);

    loss_k<<<1, 256, 0, stream>>>(Gss, Gst, cnt_s, cnt_t, nold, outp);
}